// OmegaVisionNet_19335942767320
// MI455X (gfx1250) — compile-verified
//
#include <hip/hip_runtime.h>

// Problem constants (match the reference)
#define Bsz   8
#define Cch   1024
#define Hh    56
#define Ww    56
#define NBINS 256
#define HW    (Hh * Ww)     // 3136
#define BC    (Bsz * Cch)   // 8192
#define T_THRESH 0.9f

typedef __attribute__((ext_vector_type(2))) float v2f;
typedef __attribute__((ext_vector_type(8))) float v8f;

// ---------------------------------------------------------------------------
// Workspace layout (bytes):
//   [0)                      hist/integ  u32[HW*NBINS]   (counts -> float bits in-place)
//   [HIST_BYTES)             pooled      f32[BC]
//   [HIST_BYTES + POOL_B)    scalars     u32[8]  {xd, ys, yd, area_bits}
// ---------------------------------------------------------------------------
#define HIST_BYTES ((size_t)HW * NBINS * 4)   // 3,211,264
#define POOL_BYTES ((size_t)BC * 4)           // 32,768

// ===========================================================================
// Kernel 1: per-cell histograms. One block owns (row h, half-row of 28 cells).
// LDS-private histogram -> plain global stores (no global atomics needed).
// ===========================================================================
#define WSPLIT 2
#define CW     (Ww / WSPLIT)   // 28 cells per block

__global__ void hist_kernel(const float* __restrict__ x, unsigned* __restrict__ hist) {
    __shared__ unsigned lh[CW * NBINS];   // 28 KB
    const int h     = blockIdx.x / WSPLIT;
    const int wbase = (blockIdx.x % WSPLIT) * CW;

    for (int i = threadIdx.x; i < CW * NBINS; i += blockDim.x) lh[i] = 0u;
    __syncthreads();

    const float* rowbase = x + (size_t)h * Ww + wbase;
    const int total = BC * CW;   // 8192 * 28 samples for this block
    for (int j = threadIdx.x; j < total; j += blockDim.x) {
        int bc = j / CW;
        int w  = j - bc * CW;
        float v = rowbase[(size_t)bc * HW + w];
        // torch.histc semantics: ignore outside [0,1]; v==1 -> last bin
        if (v >= 0.0f && v <= 1.0f) {
            int q = (int)floorf(v * 256.0f);
            q = q < 0 ? 0 : (q > 255 ? 255 : q);
            atomicAdd(&lh[w * NBINS + q], 1u);
        }
    }
    __syncthreads();

    unsigned* out = hist + ((size_t)h * Ww + wbase) * NBINS;
    for (int i = threadIdx.x; i < CW * NBINS; i += blockDim.x) out[i] = lh[i];
}

// ===========================================================================
// Kernel 2: column entropies (h=0 row) + argmax, in-place integral histogram,
// then the EKLM greedy region growth on a single wave32.
// ===========================================================================
__device__ __forceinline__ float wave_region_entropy(const unsigned* __restrict__ integ,
                                                     int xd, int ys, int yd, int lane) {
    // region rows [0, xd), cols [ys, yd); integ holds inclusive 2D prefix (float bits)
    const int ia = ((xd - 1) * Ww + (yd - 1)) * NBINS;
    const int ib = (ys > 0) ? ((xd - 1) * Ww + (ys - 1)) * NBINS : -1;
    float hv[8];
    float hsum = 0.0f;
#pragma unroll
    for (int j = 0; j < 8; ++j) {
        int bin = lane + 32 * j;
        float a = __uint_as_float(integ[ia + bin]);
        float b = (ib >= 0) ? __uint_as_float(integ[ib + bin]) : 0.0f;
        hv[j] = a - b;
        hsum += hv[j];
    }
#pragma unroll
    for (int m = 16; m >= 1; m >>= 1) hsum += __shfl_xor(hsum, m, 32);
    float e = 0.0f;
#pragma unroll
    for (int j = 0; j < 8; ++j) {
        float p = hv[j] / hsum;
        e -= p * log2f(p + 1e-9f);
    }
#pragma unroll
    for (int m = 16; m >= 1; m >>= 1) e += __shfl_xor(e, m, 32);
    return e;   // identical value on all 32 lanes
}

__global__ void eklm_kernel(unsigned* __restrict__ hist, unsigned* __restrict__ scalars) {
    __shared__ float ent_y[Ww];
    __shared__ int s_ys0;
    const int tid = threadIdx.x;

    // ---- phase A: entropy of each (h=0, w) cell histogram; argmax -> ys0 ----
    if (tid < Ww) {
        const unsigned* col = hist + (size_t)tid * NBINS;   // (0*Ww + w)*NBINS
        float sum = 0.0f;
        for (int b = 0; b < NBINS; ++b) sum += (float)col[b];
        float e = 0.0f;
        for (int b = 0; b < NBINS; ++b) {
            float p = (float)col[b] / sum;
            e -= p * log2f(p + 1e-9f);
        }
        ent_y[tid] = e;
    }
    __syncthreads();
    if (tid == 0) {
        int best = 0; float bv = ent_y[0];
        for (int w = 1; w < Ww; ++w)
            if (ent_y[w] > bv) { bv = ent_y[w]; best = w; }   // first max
        s_ys0 = best;
    }
    __syncthreads();

    // ---- phase B1: cumsum over h (u32 counts -> float bits, in place) ----
    // chain id c in [0, Ww*NBINS): w = c/NBINS, bin = c%NBINS
    for (int c = tid; c < Ww * NBINS; c += blockDim.x) {
        int w = c >> 8, bin = c & 255;
        float run = 0.0f;
        for (int h = 0; h < Hh; ++h) {
            int idx = (h * Ww + w) * NBINS + bin;
            run += (float)hist[idx];
            hist[idx] = __float_as_uint(run);
        }
    }
    __syncthreads();
    // ---- phase B2: cumsum over w (float bits in place) ----
    for (int c = tid; c < Hh * NBINS; c += blockDim.x) {
        int h = c >> 8, bin = c & 255;
        float run = 0.0f;
        for (int w = 0; w < Ww; ++w) {
            int idx = (h * Ww + w) * NBINS + bin;
            run += __uint_as_float(hist[idx]);
            hist[idx] = __float_as_uint(run);
        }
    }
    __syncthreads();

    // ---- phase C: greedy growth on wave 0 (uniform control flow) ----
    if (tid < 32) {
        const int lane = tid;
        const float total_ent = wave_region_entropy(hist, Hh, 0, Ww, lane);
        int xd = 1, ys = s_ys0, yd = s_ys0 + 1;
        float Ts = wave_region_entropy(hist, xd, ys, yd, lane) / total_ent;
        bool done = false;
        for (int it = 0; it < 4096 && Ts < T_THRESH && !done; ++it) {
            float e_cur = wave_region_entropy(hist, xd, ys, yd, lane);
            bool c1 = (xd + 1 < Hh) && (wave_region_entropy(hist, xd + 1, ys, yd, lane) > e_cur);
            bool c2 = !c1 && (ys - 1 >= 0) &&
                      (wave_region_entropy(hist, xd, ys - 1, yd, lane) > e_cur);
            bool c3 = !c1 && !c2 && (yd + 1 < Ww) &&
                      (wave_region_entropy(hist, xd, ys, yd + 1, lane) > e_cur);
            if (c1)       xd += 1;
            else if (c2)  ys -= 1;
            else if (c3)  yd += 1;
            done = !(c1 || c2 || c3);
            Ts = wave_region_entropy(hist, xd, ys, yd, lane) / total_ent;
        }
        if (lane == 0) {
            scalars[0] = (unsigned)xd;
            scalars[1] = (unsigned)ys;
            scalars[2] = (unsigned)yd;
            float area = (float)(xd * (yd - ys));
            if (area < 1.0f) area = 1.0f;
            scalars[3] = __float_as_uint(area);
        }
    }
}

// ===========================================================================
// Kernel 3: masked mean-pool. One block per (b,c) plane; region from scalars.
// ===========================================================================
__global__ void pool_kernel(const float* __restrict__ x,
                            const unsigned* __restrict__ scalars,
                            float* __restrict__ pooled) {
    __shared__ float red[256];
    const int bc = blockIdx.x;
    const int xd = (int)scalars[0];
    const int ys = (int)scalars[1];
    const int yd = (int)scalars[2];
    const float area = __uint_as_float(scalars[3]);
    const int rw = yd - ys;
    const int n  = xd * rw;
    const float* base = x + (size_t)bc * HW;

    float s = 0.0f;
    for (int r = threadIdx.x; r < n; r += blockDim.x) {
        int h = r / rw;
        int w = ys + (r - h * rw);
        s += base[h * Ww + w];
    }
    red[threadIdx.x] = s;
    __syncthreads();
    for (int m = 128; m >= 1; m >>= 1) {
        if (threadIdx.x < m) red[threadIdx.x] += red[threadIdx.x + m];
        __syncthreads();
    }
    if (threadIdx.x == 0) pooled[bc] = red[0] / area;
}

// ===========================================================================
// Kernel 4: pooled[8,1024] @ w_fc[1024,30] via V_WMMA_F32_16X16X4_F32.
// One wave32. M padded 8->16, N padded 30->32 (two 16-wide tiles), K=1024.
// Lane layouts per CDNA5 ISA 7.12.2 (f32 A 16x4 / B 4x16 / C-D 16x16).
// ===========================================================================
__global__ void gemm_kernel(const float* __restrict__ pooled,
                            const float* __restrict__ wfc,
                            float* __restrict__ out) {
    const int lane = threadIdx.x;         // 32 lanes, all active (WMMA needs full EXEC)
    const int m    = lane & 15;           // A row / B col within tile
    const int klo  = (lane >> 4) * 2;     // K sub-offset for this lane half

    v8f acc0 = {};   // classes 0..15
    v8f acc1 = {};   // classes 16..31 (30,31 padded with zero B)

    for (int kk = 0; kk < Cch; kk += 4) {
        const int k0 = kk + klo;
        v2f a, b0, b1;
        a.x = (m < 8) ? pooled[m * Cch + k0]     : 0.0f;   // M rows 8..15 are zero pad
        a.y = (m < 8) ? pooled[m * Cch + k0 + 1] : 0.0f;
        const int n1 = 16 + m;
        b0.x = wfc[k0 * 30 + m];
        b0.y = wfc[(k0 + 1) * 30 + m];
        b1.x = (n1 < 30) ? wfc[k0 * 30 + n1]       : 0.0f;
        b1.y = (n1 < 30) ? wfc[(k0 + 1) * 30 + n1] : 0.0f;

#if __has_builtin(__builtin_amdgcn_wmma_f32_16x16x4_f32)
        acc0 = __builtin_amdgcn_wmma_f32_16x16x4_f32(false, a, false, b0, (short)0, acc0,
                                                     false, false);
        acc1 = __builtin_amdgcn_wmma_f32_16x16x4_f32(false, a, false, b1, (short)0, acc1,
                                                     false, false);
#else
        // scalar fallback (should not be taken on gfx1250; keeps build green)
#pragma unroll
        for (int r = 0; r < 8; ++r) { acc0[r] += a.x * b0.x; acc1[r] += a.x * b1.x; }
#endif
    }

    // D layout: VGPR r <-> M = r + 8*(lane>=16); N = lane&15 (+16 for tile1)
    const int nlo  = lane & 15;
    const int mofs = (lane >> 4) * 8;
#pragma unroll
    for (int r = 0; r < 8; ++r) {
        int M = r + mofs;
        if (M < 8) {
            out[M * 30 + nlo] = acc0[r];              // N = 0..15
            int N1 = 16 + nlo;
            if (N1 < 30) out[M * 30 + N1] = acc1[r];  // N = 16..29
        }
    }
}

// ===========================================================================
extern "C" void kernel_launch(void* const* d_in, const int* in_sizes, int n_in,
                              void* d_out, int out_size, void* d_ws, size_t ws_size,
                              hipStream_t stream) {
    (void)in_sizes; (void)n_in; (void)out_size; (void)ws_size;
    const float* x   = (const float*)d_in[0];   // [8,1024,56,56] f32
    const float* wfc = (const float*)d_in[1];   // [1024,30] f32
    float* out = (float*)d_out;                 // [8,30] f32

    unsigned char* ws = (unsigned char*)d_ws;
    unsigned* hist    = (unsigned*)ws;                              // 3.2 MB
    float*    pooled  = (float*)(ws + HIST_BYTES);                  // 32 KB
    unsigned* scalars = (unsigned*)(ws + HIST_BYTES + POOL_BYTES);  // 32 B

    hist_kernel<<<Hh * WSPLIT, 256, 0, stream>>>(x, hist);
    eklm_kernel<<<1, 1024, 0, stream>>>(hist, scalars);
    pool_kernel<<<BC, 256, 0, stream>>>(x, scalars, pooled);
    gemm_kernel<<<1, 32, 0, stream>>>(pooled, wfc, out);
}